// Generator_74174085202278
// MI455X (gfx1250) — compile-verified
//
#include <hip/hip_runtime.h>
#include <hip/hip_bf16.h>
#include <math.h>

// ---------------------------------------------------------------------------
// Problem constants (from reference)
// ---------------------------------------------------------------------------
#define NB      64          // batch
#define LEN     1024        // sequence length
#define CIN     514
#define NOISEC  64
#define COMP    16
#define NCLS    3
#define MAINC   80          // COMP + NOISEC
#define C1      32          // CIN/16
#define C3      48          // NC*16
#define CH4     12          // NC*4
#define ROWS    (NB*LEN)    // 65536
#define NLIN    3072        // NC*L
#define KLIN1   12288       // CH4*L
#define KSPLIT  16

typedef __attribute__((ext_vector_type(2))) float v2f;
typedef __attribute__((ext_vector_type(8))) float v8f;

// V_WMMA_F32_16X16X4_F32 : D(16x16 f32) = A(16x4 f32) x B(4x16 f32) + C
// A/B: 2 VGPRs/lane (lanes 0-15: K=k,k+1 ; lanes 16-31: K=k+2,k+3)
// C/D: 8 VGPRs/lane (lane<16: D[v][lane], lane>=16: D[v+8][lane-16])
__device__ __forceinline__ v8f wmma4(v2f a, v2f b, v8f c) {
    return __builtin_amdgcn_wmma_f32_16x16x4_f32(
        /*neg_a=*/false, a, /*neg_b=*/false, b,
        /*c_mod=*/(short)0, c, /*reuse_a=*/false, /*reuse_b=*/false);
}

__device__ __forceinline__ float lrelu(float x) { return x > 0.0f ? x : 0.2f * x; }

// ---------------------------------------------------------------------------
// conv1: pointwise 514->32 over (N*L) rows; implicit GEMM + lrelu
// X: (ROWS, 514) row-major (== cond_data (N,L,514))
// W: ew1 (32, 514)  -> B[k][o] = W[o*514 + k]
// Y: (ROWS, 32)
// Main loop covers K=0..511 unconditionally; K=512..513 handled by a single
// masked tail step (keeps the hot loop free of EXEC-mask branching).
// ---------------------------------------------------------------------------
__global__ __launch_bounds__(256) void conv1_kernel(
    const float* __restrict__ X, const float* __restrict__ W,
    const float* __restrict__ bias, float* __restrict__ Y)
{
    const int wave = threadIdx.x >> 5, lane = threadIdx.x & 31;
    const int lm = lane & 15, half = lane >> 4;
    const int t = blockIdx.x * 8 + wave;        // 8192 tiles total
    const int tile_n = t & 1;
    const int tile_m = t >> 1;
    const int r0 = tile_m * 16, n0 = tile_n * 16;
    const int o = n0 + lm;
    const float* Arow = X + (size_t)(r0 + lm) * CIN;
    const float* Bcol = W + (size_t)o * CIN;
    v8f acc = {};
    for (int k = 0; k < 512; k += 4) {          // unconditional main body
        const int kb = k + 2 * half;
        v2f a, b;
        a.x = Arow[kb];     a.y = Arow[kb + 1];
        b.x = Bcol[kb];     b.y = Bcol[kb + 1];
        acc = wmma4(a, b, acc);
    }
    {                                           // tail: K = 512,513 valid only
        v2f a = {}, b = {};
        if (half == 0) {
            a.x = Arow[512]; a.y = Arow[513];
            b.x = Bcol[512]; b.y = Bcol[513];
        }
        acc = wmma4(a, b, acc);
    }
    const float bv = bias[o];
    #pragma unroll
    for (int v = 0; v < 8; ++v) {
        const int row = r0 + v + 8 * half;
        Y[(size_t)row * C1 + o] = lrelu(acc[v] + bv);
    }
}

// ---------------------------------------------------------------------------
// conv2: k=3, 32->16, implicit GEMM (K = 3*32 = 96), NO activation (pre-BN)
// Y1: (ROWS, 32) ; W: ew2 (16,32,3) ; Out: channel-planar (N,16,L)
// SAFE=true  -> tile is interior in l, no bounds checks on A loads.
// ---------------------------------------------------------------------------
template <bool SAFE>
__device__ __forceinline__ void conv2_acc(
    v8f& acc, const float* __restrict__ Y1, const float* __restrict__ W,
    int r, int l, int o, int half)
{
    for (int tap = 0; tap < 3; ++tap) {
        const int lt = l + tap - 1;
        const bool inb = SAFE ? true : ((unsigned)lt < (unsigned)LEN);
        const float* Ar = Y1 + (size_t)(r + tap - 1) * C1;
        for (int c = 0; c < C1; c += 4) {
            const int cc = c + 2 * half;
            v2f a, b;
            if (SAFE || inb) { a.x = Ar[cc]; a.y = Ar[cc + 1]; }
            else             { a.x = 0.0f;   a.y = 0.0f; }
            b.x = W[o * (C1 * 3) + cc * 3 + tap];
            b.y = W[o * (C1 * 3) + (cc + 1) * 3 + tap];
            acc = wmma4(a, b, acc);
        }
    }
}

__global__ __launch_bounds__(256) void conv2_kernel(
    const float* __restrict__ Y1, const float* __restrict__ W,
    const float* __restrict__ bias, float* __restrict__ C2)
{
    const int wave = threadIdx.x >> 5, lane = threadIdx.x & 31;
    const int lm = lane & 15, half = lane >> 4;
    const int tile_m = blockIdx.x * 8 + wave;   // 4096 tiles, single n-tile
    const int r0 = tile_m * 16;
    const int l0 = r0 & (LEN - 1);              // wave-uniform tile start in l
    const int r = r0 + lm;
    const int l = r & (LEN - 1);
    const int o = lm;                            // 16 output channels
    v8f acc = {};
    if (l0 != 0 && l0 != (LEN - 16)) conv2_acc<true >(acc, Y1, W, r, l, o, half);
    else                             conv2_acc<false>(acc, Y1, W, r, l, o, half);
    const float bv = bias[o];
    #pragma unroll
    for (int v = 0; v < 8; ++v) {
        const int row = r0 + v + 8 * half;       // (n,l) position
        const int n = row >> 10, ll = row & (LEN - 1);
        C2[((size_t)(n * COMP + o)) * LEN + ll] = acc[v] + bv;
    }
}

// ---------------------------------------------------------------------------
// Batch-norm statistics: one block per channel; planar input (N, C, L)
// stats[c] = mean, stats[C + c] = rsqrt(var + eps)
// ---------------------------------------------------------------------------
__global__ void bn_stats_kernel(const float* __restrict__ Xp, int C,
                                float* __restrict__ stats)
{
    __shared__ float s1[256], s2[256];
    const int c = blockIdx.x;
    float s = 0.0f, q = 0.0f;
    for (int i = threadIdx.x; i < ROWS; i += blockDim.x) {
        const int n = i >> 10, l = i & (LEN - 1);
        const float v = Xp[((size_t)(n * C + c)) * LEN + l];
        s += v; q += v * v;
    }
    s1[threadIdx.x] = s; s2[threadIdx.x] = q;
    __syncthreads();
    for (int st = 128; st > 0; st >>= 1) {
        if ((int)threadIdx.x < st) {
            s1[threadIdx.x] += s1[threadIdx.x + st];
            s2[threadIdx.x] += s2[threadIdx.x + st];
        }
        __syncthreads();
    }
    if (threadIdx.x == 0) {
        const float mean = s1[0] * (1.0f / (float)ROWS);
        const float var  = s2[0] * (1.0f / (float)ROWS) - mean * mean;
        stats[c]     = mean;
        stats[C + c] = rsqrtf(var + 1e-5f);
    }
}

// ---------------------------------------------------------------------------
// Apply BN to conv2 output -> cond (written to d_out, planar (N,16,L)), and
// build gi = concat(cond, noise^T) + positional-encoding.
// gi is stored INTERLEAVED (N, L, 80) so conv3's A fragments are contiguous.
// ---------------------------------------------------------------------------
__global__ void build_gi_kernel(
    const float* __restrict__ C2, const float* __restrict__ stats,
    const float* __restrict__ eg, const float* __restrict__ ebt,
    const float* __restrict__ noise, float* __restrict__ gi,
    float* __restrict__ cond_out)
{
    const int idx = blockIdx.x * blockDim.x + threadIdx.x;  // N*L*80
    if (idx >= NB * LEN * MAINC) return;
    const int c   = idx % MAINC;
    const int pos = idx / MAINC;                 // n*LEN + l
    const int l = pos & (LEN - 1);
    const int n = pos >> 10;
    float v;
    if (c < COMP) {
        const float x = C2[((size_t)(n * COMP + c)) * LEN + l];
        v = (x - stats[c]) * stats[COMP + c] * eg[c] + ebt[c];
        cond_out[((size_t)(n * COMP + c)) * LEN + l] = v;
    } else {
        v = noise[(size_t)pos * NOISEC + (c - COMP)];
    }
    const int d2 = c >> 1;
    const float div = __powf(10000.0f, -2.0f * (float)d2 / (float)MAINC);
    const float ang = div * (float)l;
    const float pe = (c & 1) ? __cosf(ang) : __sinf(ang);
    gi[idx] = v + pe;                            // (N, L, 80) interleaved
}

// ---------------------------------------------------------------------------
// conv3: k=7, 80->48, implicit GEMM (K = 7*80 = 560) + lrelu
// gi interleaved (N,L,80) ; W: w3 (48,80,7) ; Out: (ROWS, 48)
// ---------------------------------------------------------------------------
template <bool SAFE>
__device__ __forceinline__ void conv3_acc(
    v8f& acc, const float* __restrict__ gi, const float* __restrict__ Wo,
    int r, int l, int half)
{
    for (int tap = 0; tap < 7; ++tap) {
        const int lt = l + tap - 3;
        const bool inb = SAFE ? true : ((unsigned)lt < (unsigned)LEN);
        const float* Ar = gi + (size_t)(r + tap - 3) * MAINC;  // row (n, lt)
        for (int c = 0; c < MAINC; c += 4) {
            const int cc = c + 2 * half;
            v2f a, b;
            if (SAFE || inb) { a.x = Ar[cc]; a.y = Ar[cc + 1]; }
            else             { a.x = 0.0f;   a.y = 0.0f; }
            b.x = Wo[cc * 7 + tap];
            b.y = Wo[(cc + 1) * 7 + tap];
            acc = wmma4(a, b, acc);
        }
    }
}

__global__ __launch_bounds__(256) void conv3_kernel(
    const float* __restrict__ gi, const float* __restrict__ W,
    const float* __restrict__ bias, float* __restrict__ H3)
{
    const int wave = threadIdx.x >> 5, lane = threadIdx.x & 31;
    const int lm = lane & 15, half = lane >> 4;
    const int t = blockIdx.x * 8 + wave;         // 12288 tiles
    const int tile_n = t % 3;
    const int tile_m = t / 3;
    const int r0 = tile_m * 16, n0 = tile_n * 16;
    const int l0 = r0 & (LEN - 1);               // wave-uniform
    const int r = r0 + lm;
    const int l = r & (LEN - 1);
    const int o = n0 + lm;
    const float* Wo = W + (size_t)o * (MAINC * 7);
    v8f acc = {};
    if (l0 != 0 && l0 != (LEN - 16)) conv3_acc<true >(acc, gi, Wo, r, l, half);
    else                             conv3_acc<false>(acc, gi, Wo, r, l, half);
    const float bv = bias[o];
    #pragma unroll
    for (int v = 0; v < 8; ++v) {
        const int row = r0 + v + 8 * half;
        H3[(size_t)row * C3 + o] = lrelu(acc[v] + bv);
    }
}

// ---------------------------------------------------------------------------
// conv4: k=3, 48->12, implicit GEMM (K = 3*48 = 144), pre-BN output
// H3: (ROWS, 48) ; W: w4 (12,48,3) ; Out planar (N, 12, L)
// N-dim uses one 16-wide tile; columns 12..15 are masked.
// ---------------------------------------------------------------------------
template <bool SAFE>
__device__ __forceinline__ void conv4_acc(
    v8f& acc, const float* __restrict__ H3, const float* __restrict__ W,
    int r, int l, int o, bool ov, int half)
{
    for (int tap = 0; tap < 3; ++tap) {
        const int lt = l + tap - 1;
        const bool inb = SAFE ? true : ((unsigned)lt < (unsigned)LEN);
        const float* Ar = H3 + (size_t)(r + tap - 1) * C3;
        for (int c = 0; c < C3; c += 4) {
            const int cc = c + 2 * half;
            v2f a, b;
            if (SAFE || inb) { a.x = Ar[cc]; a.y = Ar[cc + 1]; }
            else             { a.x = 0.0f;   a.y = 0.0f; }
            b.x = ov ? W[o * (C3 * 3) + cc * 3 + tap]       : 0.0f;
            b.y = ov ? W[o * (C3 * 3) + (cc + 1) * 3 + tap] : 0.0f;
            acc = wmma4(a, b, acc);
        }
    }
}

__global__ __launch_bounds__(256) void conv4_kernel(
    const float* __restrict__ H3, const float* __restrict__ W,
    const float* __restrict__ bias, float* __restrict__ C4out)
{
    const int wave = threadIdx.x >> 5, lane = threadIdx.x & 31;
    const int lm = lane & 15, half = lane >> 4;
    const int tile_m = blockIdx.x * 8 + wave;    // 4096 tiles
    const int r0 = tile_m * 16;
    const int l0 = r0 & (LEN - 1);               // wave-uniform
    const int r = r0 + lm;
    const int l = r & (LEN - 1);
    const int o = lm;
    const bool ov = (o < CH4);
    v8f acc = {};
    if (l0 != 0 && l0 != (LEN - 16)) conv4_acc<true >(acc, H3, W, r, l, o, ov, half);
    else                             conv4_acc<false>(acc, H3, W, r, l, o, ov, half);
    if (ov) {
        const float bv = bias[o];
        #pragma unroll
        for (int v = 0; v < 8; ++v) {
            const int row = r0 + v + 8 * half;
            const int n = row >> 10, ll = row & (LEN - 1);
            C4out[((size_t)(n * CH4 + o)) * LEN + ll] = acc[v] + bv;
        }
    }
}

// ---------------------------------------------------------------------------
// BN + lrelu over conv4 output -> h (N, 12288) row-major (== planar (N,12,L))
// ---------------------------------------------------------------------------
__global__ void bn_lrelu_kernel(const float* __restrict__ C4a,
                                const float* __restrict__ stats,
                                const float* __restrict__ g,
                                const float* __restrict__ bt,
                                float* __restrict__ H)
{
    const int idx = blockIdx.x * blockDim.x + threadIdx.x;  // N*12*L
    if (idx >= NB * CH4 * LEN) return;
    const int c = (idx >> 10) % CH4;
    const float x = C4a[idx];
    const float v = (x - stats[c]) * stats[CH4 + c] * g[c] + bt[c];
    H[idx] = lrelu(v);
}

// ---------------------------------------------------------------------------
// K-split GEMM: part[ks][0..63][n] = A(64 x K)[., k0:k0+Kc] @ B(K x 3072)
// One wave per 16-wide N tile; the wave carries all four 16-row M tiles so
// each B element is fetched from HBM exactly once (B is the dominant traffic).
// grid = (Ntiles/8, KSPLIT), block = 256.
// ---------------------------------------------------------------------------
__global__ __launch_bounds__(256) void gemm_ksplit_kernel(
    const float* __restrict__ A, const float* __restrict__ B,
    float* __restrict__ part, int K, int Kc)
{
    const int wave = threadIdx.x >> 5, lane = threadIdx.x & 31;
    const int lm = lane & 15, half = lane >> 4;
    const int tile_n = blockIdx.x * 8 + wave;
    const int n0 = tile_n * 16;
    const int coln = n0 + lm;
    const int k0 = blockIdx.y * Kc;
    const int kend = k0 + Kc;
    const float* Bp = B + coln;
    const float* A0 = A + (size_t)lm * K;
    const float* A1 = A0 + (size_t)16 * K;
    const float* A2 = A0 + (size_t)32 * K;
    const float* A3 = A0 + (size_t)48 * K;
    v8f acc0 = {}, acc1 = {}, acc2 = {}, acc3 = {};
    for (int k = k0; k < kend; k += 4) {
        const int kb = k + 2 * half;
        __builtin_prefetch(&Bp[(size_t)(kb + 16) * NLIN], 0, 1);
        v2f b, a;
        b.x = Bp[(size_t)kb * NLIN];
        b.y = Bp[(size_t)(kb + 1) * NLIN];
        a.x = A0[kb]; a.y = A0[kb + 1];
        acc0 = wmma4(a, b, acc0);
        a.x = A1[kb]; a.y = A1[kb + 1];
        acc1 = wmma4(a, b, acc1);
        a.x = A2[kb]; a.y = A2[kb + 1];
        acc2 = wmma4(a, b, acc2);
        a.x = A3[kb]; a.y = A3[kb + 1];
        acc3 = wmma4(a, b, acc3);
    }
    float* pp = part + (size_t)blockIdx.y * 64 * NLIN;
    #pragma unroll
    for (int v = 0; v < 8; ++v) {
        const int row = v + 8 * half;
        pp[(size_t)(row)      * NLIN + coln] = acc0[v];
        pp[(size_t)(row + 16) * NLIN + coln] = acc1[v];
        pp[(size_t)(row + 32) * NLIN + coln] = acc2[v];
        pp[(size_t)(row + 48) * NLIN + coln] = acc3[v];
    }
}

// Reduce K-split partials + bias (+ optional lrelu)
__global__ void ksplit_reduce_kernel(const float* __restrict__ part,
                                     const float* __restrict__ bias,
                                     float* __restrict__ out, int do_lrelu)
{
    const int idx = blockIdx.x * blockDim.x + threadIdx.x;  // 64*3072
    if (idx >= 64 * NLIN) return;
    const int col = idx % NLIN;
    float s = bias[col];
    #pragma unroll
    for (int i = 0; i < KSPLIT; ++i) s += part[(size_t)i * 64 * NLIN + idx];
    out[idx] = do_lrelu ? lrelu(s) : s;
}

// ---------------------------------------------------------------------------
// label revise + gumbel softmax + straight-through one-hot -> gen
// logits: (N, L, 3)
// ---------------------------------------------------------------------------
__device__ __forceinline__ int argmax3(float a, float b, float c) {
    int i = 0; float m = a;
    if (b > m) { m = b; i = 1; }
    if (c > m) { i = 2; }
    return i;
}

__global__ void gen_kernel(const float* __restrict__ logits,
                           const float* __restrict__ gumbel,
                           float* __restrict__ gen)
{
    const int idx = blockIdx.x * blockDim.x + threadIdx.x;  // N*L
    if (idx >= ROWS) return;
    const int l = idx & (LEN - 1);
    const float* row = logits + (size_t)idx * NCLS;
    float r0 = row[0], r1 = row[1], r2 = row[2];

    const int base = idx & ~3;                       // 4-block start (same n)
    const float* rb = logits + (size_t)base * NCLS;
    const int predm = argmax3(rb[0], rb[1], rb[2]);
    int nxt = -1;
    if ((l >> 2) != (LEN / 4 - 1)) {
        const float* rn = logits + (size_t)(base + 4) * NCLS;
        nxt = argmax3(rn[0], rn[1], rn[2]);
    }
    const bool chain = (predm == 2) && (nxt == 2);
    const int phase = l & 3;
    int prev = 0;
    if (l > 0) {
        const float* rp = logits + (size_t)(idx - 1) * NCLS;
        prev = argmax3(rp[0], rp[1], rp[2]);
    }
    const bool c100 = (((phase == 1) && (predm == 1 || predm == 2)) ||
                       ((phase == 3) && (prev == 1))) && !chain;
    if (c100)                { r0 = 1.0f; r1 = 0.0f; r2 = 0.0f; }
    if (chain && phase != 0) { r0 = 0.0f; r1 = 0.0f; r2 = 1.0f; }

    const float g0 = r0 + gumbel[(size_t)idx * 3 + 0];
    const float g1 = r1 + gumbel[(size_t)idx * 3 + 1];
    const float g2 = r2 + gumbel[(size_t)idx * 3 + 2];
    const float mx = fmaxf(g0, fmaxf(g1, g2));
    const float e0 = __expf(g0 - mx), e1 = __expf(g1 - mx), e2 = __expf(g2 - mx);
    const float inv = 1.0f / (e0 + e1 + e2);
    const float s0 = e0 * inv, s1 = e1 * inv, s2 = e2 * inv;
    const int am = argmax3(s0, s1, s2);
    const float h0 = (am == 0) ? 1.0f : 0.0f;
    const float h1 = (am == 1) ? 1.0f : 0.0f;
    const float h2 = (am == 2) ? 1.0f : 0.0f;
    gen[(size_t)idx * 3 + 0] = (h0 + s0) - s0;   // matches y_hard + y_soft - sg(y_soft)
    gen[(size_t)idx * 3 + 1] = (h1 + s1) - s1;
    gen[(size_t)idx * 3 + 2] = (h2 + s2) - s2;
}

// ---------------------------------------------------------------------------
// Orchestration
// ---------------------------------------------------------------------------
extern "C" void kernel_launch(void* const* d_in, const int* in_sizes, int n_in,
                              void* d_out, int out_size, void* d_ws, size_t ws_size,
                              hipStream_t stream)
{
    const float* noise     = (const float*)d_in[0];
    const float* cond_data = (const float*)d_in[1];
    const float* gumbel    = (const float*)d_in[2];
    const float* ew1 = (const float*)d_in[3];
    const float* eb1 = (const float*)d_in[4];
    const float* ew2 = (const float*)d_in[5];
    const float* eb2 = (const float*)d_in[6];
    const float* eg  = (const float*)d_in[7];
    const float* ebt = (const float*)d_in[8];
    const float* w3  = (const float*)d_in[9];
    const float* b3  = (const float*)d_in[10];
    const float* w4  = (const float*)d_in[11];
    const float* b4  = (const float*)d_in[12];
    const float* g4  = (const float*)d_in[13];
    const float* bt4 = (const float*)d_in[14];
    const float* lw1 = (const float*)d_in[15];
    const float* lb1 = (const float*)d_in[16];
    const float* lw2 = (const float*)d_in[17];
    const float* lb2 = (const float*)d_in[18];
    const float* lw3 = (const float*)d_in[19];
    const float* lb3 = (const float*)d_in[20];

    float* out_gen  = (float*)d_out;                       // (64,1024,3)
    float* out_cond = (float*)d_out + (size_t)ROWS * NCLS; // (64,16,1024)

    // workspace layout (floats)
    float* ws = (float*)d_ws;
    size_t off = 0;
    float* y1      = ws + off; off += (size_t)ROWS * C1;        // 2M
    float* c2      = ws + off; off += (size_t)NB * COMP * LEN;  // 1M
    float* stats16 = ws + off; off += 64;
    float* stats12 = ws + off; off += 64;
    float* gi      = ws + off; off += (size_t)ROWS * MAINC;     // 5.25M (N,L,80)
    float* h3      = ws + off; off += (size_t)ROWS * C3;        // 3.1M
    float* c4      = ws + off; off += (size_t)NB * CH4 * LEN;   // 0.75M
    float* hA      = ws + off; off += (size_t)NB * CH4 * LEN;   // 0.75M
    float* part    = ws + off; off += (size_t)KSPLIT * 64 * NLIN; // 3.1M
    float* hB      = ws + off; off += (size_t)64 * NLIN;
    float* hC      = ws + off; off += (size_t)64 * NLIN;
    float* logits  = ws + off; off += (size_t)64 * NLIN;

    // 1) conv1 (1x1, 514->32) + lrelu   [8192 wave tiles]
    conv1_kernel<<<1024, 256, 0, stream>>>(cond_data, ew1, eb1, y1);

    // 2) conv2 (k=3, 32->16), pre-BN    [4096 wave tiles]
    conv2_kernel<<<512, 256, 0, stream>>>(y1, ew2, eb2, c2);

    // 3) BN stats over (N,L) per channel
    bn_stats_kernel<<<COMP, 256, 0, stream>>>(c2, COMP, stats16);

    // 4) apply BN -> cond (to d_out) and build gi (+pos encoding)
    {
        const int n = ROWS * MAINC;
        build_gi_kernel<<<(n + 255) / 256, 256, 0, stream>>>(
            c2, stats16, eg, ebt, noise, gi, out_cond);
    }

    // 5) conv3 (k=7, 80->48) + lrelu    [12288 wave tiles]
    conv3_kernel<<<1536, 256, 0, stream>>>(gi, w3, b3, h3);

    // 6) conv4 (k=3, 48->12), pre-BN    [4096 wave tiles]
    conv4_kernel<<<512, 256, 0, stream>>>(h3, w4, b4, c4);

    // 7) BN stats + 8) BN + lrelu -> h (64 x 12288)
    bn_stats_kernel<<<CH4, 256, 0, stream>>>(c4, CH4, stats12);
    {
        const int n = NB * CH4 * LEN;
        bn_lrelu_kernel<<<(n + 255) / 256, 256, 0, stream>>>(c4, stats12, g4, bt4, hA);
    }

    // 9) lin1: (64 x 12288) @ (12288 x 3072) + lrelu
    {
        dim3 grid(NLIN / 16 / 8, KSPLIT);   // (24, 16)
        gemm_ksplit_kernel<<<grid, 256, 0, stream>>>(hA, lw1, part, KLIN1, KLIN1 / KSPLIT);
        const int n = 64 * NLIN;
        ksplit_reduce_kernel<<<(n + 255) / 256, 256, 0, stream>>>(part, lb1, hB, 1);
    }
    // 10) lin2: (64 x 3072) @ (3072 x 3072) + lrelu
    {
        dim3 grid(NLIN / 16 / 8, KSPLIT);
        gemm_ksplit_kernel<<<grid, 256, 0, stream>>>(hB, lw2, part, NLIN, NLIN / KSPLIT);
        const int n = 64 * NLIN;
        ksplit_reduce_kernel<<<(n + 255) / 256, 256, 0, stream>>>(part, lb2, hC, 1);
    }
    // 11) lin3: (64 x 3072) @ (3072 x 3072) -> logits
    {
        dim3 grid(NLIN / 16 / 8, KSPLIT);
        gemm_ksplit_kernel<<<grid, 256, 0, stream>>>(hC, lw3, part, NLIN, NLIN / KSPLIT);
        const int n = 64 * NLIN;
        ksplit_reduce_kernel<<<(n + 255) / 256, 256, 0, stream>>>(part, lb3, logits, 0);
    }

    // 12) label revise + gumbel softmax + one-hot -> gen
    gen_kernel<<<ROWS / 256, 256, 0, stream>>>(logits, gumbel, out_gen);
}